// PCEModel_62337155334338
// MI455X (gfx1250) — compile-verified
//
#include <hip/hip_runtime.h>

typedef float v2f __attribute__((ext_vector_type(2)));
typedef float v8f __attribute__((ext_vector_type(8)));

#define MAXT 8  // max nonzero exponents per basis row (total order <= 8; reference uses P=2)

// ---------------------------------------------------------------------------
// Kernel 1: dense idxset rows (nbasis x D) -> packed sparse terms.
// code = (dim << 4) | ord, ascending-dim order (matches reference scan order;
// skipped ord=0 factors are exactly 1.0f so the product is bit-identical).
// ---------------------------------------------------------------------------
__global__ void pce_sparsify_kernel(const int* __restrict__ idxset, int nbasis, int D,
                                    int* __restrict__ cnt, int* __restrict__ codes) {
    int b = blockIdx.x * blockDim.x + threadIdx.x;
    if (b >= nbasis) return;
    int c = 0;
    for (int j = 0; j < D; ++j) {
        int o = idxset[(size_t)b * D + j];
        if (o != 0 && c < MAXT) {
            codes[(size_t)b * MAXT + c] = (j << 4) | (o & 15);
            ++c;
        }
    }
    cnt[b] = c;
}

// ---------------------------------------------------------------------------
// Kernel 2: polyval via WMMA.  pv[(d*p1+ord)*N + n] = sum_k B[ord,k] * xn^k
// One wave per (feature dim d) x (16 samples) WMMA:
//   A (16x4 f32) = Hermite coeffs, row=ord (zero-padded), K=power index
//   B (4x16 f32) = powers of xn, column = sample
//   D tile: lanes 0-15 hold column `lane`, VGPR r holds row M=r -> pv[ord][n]
// ---------------------------------------------------------------------------
__global__ void pce_pv_wmma_kernel(const float* __restrict__ x,
                                   const float* __restrict__ mean,
                                   const float* __restrict__ var,
                                   const float* __restrict__ basis,   // p1 x p1
                                   float* __restrict__ pv,            // (D*p1) x N
                                   int N, int D, int p1) {
    extern __shared__ float xs[];              // D * 16 normalized samples
    const int n0  = blockIdx.x * 16;
    const int tid = threadIdx.x;

    // staging: per sample row, threads stride over dims (coalesced, no int div)
    for (int s = 0; s < 16; ++s) {
        const int n = n0 + s;
        for (int d = tid; d < D; d += blockDim.x) {
            float v = 0.0f;
            if (n < N) v = (x[(size_t)n * D + d] - mean[d]) / var[d];
            xs[(d << 4) + s] = v;
        }
    }
    __syncthreads();

    const int lane   = tid & 31;
    const int wave   = tid >> 5;
    const int nwaves = blockDim.x >> 5;
    const int half   = lane >> 4;   // 0: K in {0,1}; 1: K in {2,3}
    const int sloc   = lane & 15;   // A row / B column within the tile

    // A fragment: basis coefficients (invariant across the d loop)
    v2f afrag;
#pragma unroll
    for (int v = 0; v < 2; ++v) {
        int row = sloc;
        int k   = v + 2 * half;
        float a = 0.0f;
        if (row < p1 && k < p1) a = basis[row * p1 + k];
        afrag[v] = a;
    }

    for (int d = wave; d < D; d += nwaves) {   // wave-uniform: EXEC all-1s at WMMA
        float xv = xs[(d << 4) + sloc];
        v2f bfrag;
#pragma unroll
        for (int v = 0; v < 2; ++v) {
            int k = v + 2 * half;
            float p = 1.0f;                    // finite powers; zero A coeffs kill k>=p1
            if (k >= 1) p = xv;
            if (k >= 2) p *= xv;
            if (k >= 3) p *= xv;
            bfrag[v] = p;
        }
        v8f c = {};
        c = __builtin_amdgcn_wmma_f32_16x16x4_f32(false, afrag, false, bfrag,
                                                  (short)0, c, false, false);
        if (lane < 16) {                       // low lanes hold rows (ord) 0..7
            int n = n0 + lane;
            if (n < N) {
#pragma unroll 4
                for (int ord = 0; ord < p1 && ord < 8; ++ord)
                    pv[(size_t)(d * p1 + ord) * N + n] = c[ord];
            }
        }
    }
}

// ---------------------------------------------------------------------------
// Kernel 3: Phi[n,b] = prod of pv terms.  Block owns 32 samples; polyval slice
// staged in LDS (stride 33, bank-conflict free) via async global->LDS b128
// copies (ASYNCcnt), no VGPR round-trip.  Lane -> b so all output stores are
// fully coalesced 128B transactions.
// ---------------------------------------------------------------------------
__global__ void pce_phi_kernel(const float* __restrict__ pv,
                               const int* __restrict__ cnt,
                               const int* __restrict__ codes,
                               float* __restrict__ out,
                               int N, int D, int p1, int nbasis) {
    extern __shared__ float tile[];               // (D*p1) rows x 33 floats
    const int R   = D * p1;
    const int n0  = blockIdx.x * 32;
    const int tid = threadIdx.x;

    if (n0 + 31 < N) {
        // full tile: async copy, 16B per lane, 8 chunks per 32-float row
        for (int j = tid; j < R * 8; j += blockDim.x) {
            const int r  = j >> 3;
            const int ci = (j & 7) << 2;          // element offset within row
            const float* g = pv + (size_t)r * N + n0 + ci;
            unsigned int lds_off =
                (unsigned int)(unsigned long long)(tile + r * 33 + ci);
            asm volatile("global_load_async_to_lds_b128 %0, %1, off"
                         :: "v"(lds_off), "v"(g) : "memory");
        }
        asm volatile("s_wait_asynccnt 0x0" ::: "memory");
    } else {
        // tail tile: scalar staging with bounds checks
        for (int j = tid; j < R * 32; j += blockDim.x) {
            int r = j >> 5, i = j & 31;
            int n = n0 + i;
            tile[r * 33 + i] = (n < N) ? pv[(size_t)r * N + n] : 0.0f;
        }
    }
    __syncthreads();

    const int lane   = tid & 31;
    const int wave   = tid >> 5;
    const int nwaves = blockDim.x >> 5;
    const int nchunk = (nbasis + 31) >> 5;

    for (int ch = wave; ch < nchunk; ch += nwaves) {
        int b = ch * 32 + lane;
        int c = 0;
        int roff[MAXT];
        if (b < nbasis) {
            c = cnt[b];
#pragma unroll
            for (int t = 0; t < MAXT; ++t) {
                if (t < c) {
                    int code = codes[(size_t)b * MAXT + t];
                    int dim  = code >> 4;
                    int ord  = code & 15;
                    roff[t]  = (dim * p1 + ord) * 33;
                }
            }
        }
        for (int i = 0; i < 32; ++i) {
            int n = n0 + i;
            if (n >= N) break;
            float acc = 1.0f;
            for (int t = 0; t < c; ++t) acc *= tile[roff[t] + i];
            if (b < nbasis) out[(size_t)n * nbasis + b] = acc;
        }
    }
}

// ---------------------------------------------------------------------------
extern "C" void kernel_launch(void* const* d_in, const int* in_sizes, int n_in,
                              void* d_out, int out_size, void* d_ws, size_t ws_size,
                              hipStream_t stream) {
    const float* x      = (const float*)d_in[0];
    const float* mean   = (const float*)d_in[1];
    const float* var    = (const float*)d_in[2];
    const float* basis  = (const float*)d_in[3];
    const int*   idxset = (const int*)d_in[4];

    const int D = in_sizes[1];
    int p1 = 1;
    while (p1 * p1 < in_sizes[3]) ++p1;           // oneDbasis is (p1, p1)
    const int N      = in_sizes[0] / D;
    const int nbasis = in_sizes[4] / D;
    float* out = (float*)d_out;

    // workspace: pv (D*p1*N floats) | cnt (nbasis ints) | codes (nbasis*MAXT ints)
    char*  ws      = (char*)d_ws;
    float* pv      = (float*)ws;
    size_t pvBytes = (size_t)D * p1 * N * sizeof(float);
    int*   cnt     = (int*)(ws + pvBytes);
    int*   codes   = cnt + nbasis;

    pce_sparsify_kernel<<<(nbasis + 255) / 256, 256, 0, stream>>>(idxset, nbasis, D, cnt, codes);

    const int tiles16 = (N + 15) / 16;
    pce_pv_wmma_kernel<<<tiles16, 256, (size_t)D * 16 * sizeof(float), stream>>>(
        x, mean, var, basis, pv, N, D, p1);

    const int tiles32 = (N + 31) / 32;
    pce_phi_kernel<<<tiles32, 256, (size_t)D * p1 * 33 * sizeof(float), stream>>>(
        pv, cnt, codes, out, N, D, p1, nbasis);
}